// FSRS3_34943853920362
// MI455X (gfx1250) — compile-verified
//
#include <hip/hip_runtime.h>

#define SEQN  64
#define BLOCK 256
#define DEPTH 8   // async stages in flight per wave

typedef float v2f __attribute__((ext_vector_type(2)));

// ---- fast hardware transcendentals (raw v_exp_f32 / v_log_f32 / v_rcp_f32) ----
__device__ __forceinline__ float fexp2(float x) {
#if defined(__HIP_DEVICE_COMPILE__) && __has_builtin(__builtin_amdgcn_exp2f)
  return __builtin_amdgcn_exp2f(x);
#else
  return exp2f(x);
#endif
}
__device__ __forceinline__ float flog2(float x) {
#if defined(__HIP_DEVICE_COMPILE__) && __has_builtin(__builtin_amdgcn_logf)
  return __builtin_amdgcn_logf(x);
#else
  return log2f(x);
#endif
}
__device__ __forceinline__ float frcp(float x) {
#if defined(__HIP_DEVICE_COMPILE__) && __has_builtin(__builtin_amdgcn_rcpf)
  return __builtin_amdgcn_rcpf(x);
#else
  return 1.0f / x;
#endif
}

// ---- CDNA5 async global->LDS copy (gfx1250), per-lane 8B, tracked by ASYNCcnt ----
__device__ __forceinline__ void async_load_b64(const float* gaddr, unsigned lds_byte_addr) {
#if defined(__HIP_DEVICE_COMPILE__)
  asm volatile("global_load_async_to_lds_b64 %0, %1, off"
               :
               : "v"(lds_byte_addr), "v"(gaddr)
               : "memory");
#endif
}
__device__ __forceinline__ void wait_async_le7() {
#if defined(__HIP_DEVICE_COMPILE__)
  asm volatile("s_wait_asynccnt 0x7" ::: "memory");
#endif
}
__device__ __forceinline__ void wait_async_0() {
#if defined(__HIP_DEVICE_COMPILE__)
  asm volatile("s_wait_asynccnt 0x0" ::: "memory");
#endif
}
__device__ __forceinline__ void wait_ds_0() {
#if defined(__HIP_DEVICE_COMPILE__)
  asm volatile("s_wait_dscnt 0x0" ::: "memory");
#endif
}

__global__ __launch_bounds__(BLOCK) void FSRS3_scan_kernel(
    const float* __restrict__ in,   // (SEQ, batch, 2)
    const float* __restrict__ w,    // 13 weights
    float* __restrict__ out,        // (SEQ, batch, 2) then (batch, 2) final state
    int batch) {
  __shared__ v2f stage[DEPTH * BLOCK];   // 16 KB: 8 stages x 256 lanes x float2

  const int tid = threadIdx.x;
  const unsigned b = blockIdx.x * BLOCK + tid;
  if (b >= (unsigned)batch) return;

  // Uniform weight loads -> scalar cache; precompute fused constants.
  const float w0 = w[0], w1 = w[1], w2 = w[2], w3 = w[3], w4 = w[4], w5 = w[5],
              w6 = w[6], w7 = w[7], w8 = w[8], w9 = w[9], w10 = w[10],
              w11 = w[11], w12 = w[12];
  const float LOG2E = 1.4426950408889634f;
  const float L09   = -0.15200309344504997f;  // log2(0.9)
  const float ew6 = fexp2(w6 * LOG2E);        // exp(w6)
  const float a_d = w5 * w2;
  const float b_d = 1.0f - w5;
  const float c8  = w8  * LOG2E;
  const float c12 = w12 * LOG2E;

  const size_t colstep = (size_t)batch * 2;        // floats per seq step
  const float* gcol = in  + (size_t)b * 2;
  float*       ocol = out + (size_t)b * 2;

  const unsigned lds0    = (unsigned)(unsigned long long)(&stage[tid]);
  const unsigned SSTRIDE = BLOCK * sizeof(v2f);    // 2048 B per stage

  // ---- prologue: fill the 8-deep async pipeline ----
#pragma unroll
  for (int k = 0; k < DEPTH; ++k)
    async_load_b64(gcol + (size_t)k * colstep, lds0 + (unsigned)k * SSTRIDE);

  float s, d;

  // ---- step 0 (peeled: init formula uses rating only) ----
  {
    wait_async_le7();
    v2f x = stage[tid];
    wait_ds_0();
    async_load_b64(gcol + (size_t)DEPTH * colstep, lds0);  // reuse stage 0 for step 8
    const float rating = x.y;
    s = fminf(fmaxf(w0 + w1 * (rating - 1.0f), 0.1f), 36500.0f);
    d = fminf(fmaxf(w2 + w3 * (rating - 3.0f), 1.0f), 10.0f);
    v2f o; o.x = s; o.y = d;
    __builtin_nontemporal_store(o, (v2f*)ocol);
  }

  // ---- main pipelined loop: k = 1 .. SEQ-DEPTH-1 ----
  for (int k = 1; k < SEQN - DEPTH; ++k) {
    wait_async_le7();                         // step-k tile has landed (in-order)
    v2f x = stage[(k & (DEPTH - 1)) * BLOCK + tid];
    wait_ds_0();                              // retire read before stage reuse
    async_load_b64(gcol + (size_t)(k + DEPTH) * colstep,
                   lds0 + (unsigned)(k & (DEPTH - 1)) * SSTRIDE);

    const float t = x.x, rating = x.y;
    const float r   = fexp2(L09 * t * frcp(s));           // 0.9^(t/s)
    const float nd  = fminf(fmaxf(a_d + b_d * (d + w4 * (rating - 3.0f)), 1.0f), 10.0f);
    const float omr = 1.0f - r;
    const float ls  = flog2(s);
    const float succ = s * (1.0f + ew6 * (11.0f - nd) * fexp2(w7 * ls)
                                 * (fexp2(c8 * omr) - 1.0f));
    const float fail = w9 * fexp2(w10 * flog2(nd) + w11 * ls + c12 * omr); // fused pow*pow*exp
    s = fminf(fmaxf(rating > 1.0f ? succ : fail, 0.1f), 36500.0f);
    d = nd;

    v2f o; o.x = s; o.y = d;
    __builtin_nontemporal_store(o, (v2f*)(ocol + (size_t)k * colstep));
  }

  // ---- tail: everything issued; drain once, then consume remaining stages ----
  wait_async_0();
#pragma unroll
  for (int k = SEQN - DEPTH; k < SEQN; ++k) {
    v2f x = stage[(k & (DEPTH - 1)) * BLOCK + tid];
    const float t = x.x, rating = x.y;
    const float r   = fexp2(L09 * t * frcp(s));
    const float nd  = fminf(fmaxf(a_d + b_d * (d + w4 * (rating - 3.0f)), 1.0f), 10.0f);
    const float omr = 1.0f - r;
    const float ls  = flog2(s);
    const float succ = s * (1.0f + ew6 * (11.0f - nd) * fexp2(w7 * ls)
                                 * (fexp2(c8 * omr) - 1.0f));
    const float fail = w9 * fexp2(w10 * flog2(nd) + w11 * ls + c12 * omr);
    s = fminf(fmaxf(rating > 1.0f ? succ : fail, 0.1f), 36500.0f);
    d = nd;

    v2f o; o.x = s; o.y = d;
    __builtin_nontemporal_store(o, (v2f*)(ocol + (size_t)k * colstep));
  }

  // ---- final_state = state after last step ----
  v2f fin; fin.x = s; fin.y = d;
  __builtin_nontemporal_store(fin, (v2f*)(out + (size_t)SEQN * colstep + (size_t)b * 2));
}

extern "C" void kernel_launch(void* const* d_in, const int* in_sizes, int n_in,
                              void* d_out, int out_size, void* d_ws, size_t ws_size,
                              hipStream_t stream) {
  (void)n_in; (void)out_size; (void)d_ws; (void)ws_size;
  const float* in = (const float*)d_in[0];
  const float* w  = (const float*)d_in[1];
  float* out = (float*)d_out;
  const int batch = in_sizes[0] / (SEQN * 2);   // 262144
  dim3 grid((batch + BLOCK - 1) / BLOCK), block(BLOCK);
  FSRS3_scan_kernel<<<grid, block, 0, stream>>>(in, w, out, batch);
}